// GNN_5214090297530
// MI455X (gfx1250) — compile-verified
//
#include <hip/hip_runtime.h>
#include <hip/hip_bf16.h>

// ---------------------------------------------------------------------------
// GIN (5-layer) forward on MI455X / gfx1250.
// All GEMMs run on V_WMMA_F32_16X16X4_F32 (full f32 precision, matches ref).
// Dims padded: EMB 300->320, 2*EMB 600->640 so N/K tiles are exact.
// Each wave computes a 32x64 tile (two 16x16x4 chains sharing B fragments).
// GEMM1 fuses BatchNorm statistics into its epilogue; GEMM2 fuses the
// BN-apply + ReLU transform into its A-fragment loads.
// ---------------------------------------------------------------------------

typedef float v2f __attribute__((ext_vector_type(2)));
typedef float v8f __attribute__((ext_vector_type(8)));

#define NN      50000
#define EE      250000
#define EMB     300
#define EMBP    320
#define D2      600
#define DP      640
#define INDIM   1024
#define NLAYER  5

// ---------------------------------------------------------------------------
// f32 WMMA GEMM:  C[M,N] = f(A)[M,K] @ B[K,N] + bias[N]
//   AMODE: 0 = A used as-is; 1 = A' = relu(A*ascl[k] + ashf[k])  (fused BN)
//   EPI:   0 = optional ReLU on output; 1 = accumulate column sum/sumsq
// Requirements: M%16==0, N%64==0, K%16==0. One wave = 32 rows x 64 cols.
// ---------------------------------------------------------------------------
template <int AMODE, int EPI>
__global__ __launch_bounds__(256)
void gemm_wmma_f32(const float* __restrict__ A, const float* __restrict__ B,
                   const float* __restrict__ bias, float* __restrict__ C,
                   const float* __restrict__ ascl, const float* __restrict__ ashf,
                   float* __restrict__ sums, float* __restrict__ sqs,
                   int M, int N, int K, int lda, int ldb, int ldc, int relu)
{
    const int lane = threadIdx.x & 31;
    const int wave = threadIdx.x >> 5;
    const int mt2 = (M + 31) >> 5;   // 32-row pair-tiles
    const int nt  = N >> 6;          // 64-col tiles
    const long tiles = (long)mt2 * nt;
    const long tile = (long)blockIdx.x * 8 + wave;
    if (tile >= tiles) return;       // wave-uniform: EXEC all-ones inside

    const int tm = (int)(tile % mt2);
    const int tn = (int)(tile / mt2);
    const int m0 = tm << 5;
    const int n0 = tn << 6;

    const int mrow = lane & 15;           // A: M index within tile
    const int kb   = (lane >> 4) << 1;    // A/B: K sub-offset (0 or 2)
    const int ncol = lane & 15;           // B/C: N index within 16-col subtile

    const int ra0 = m0 + mrow;                                  // < M always
    const int ra1t = m0 + 16 + mrow;
    const int ra1 = (ra1t < M) ? ra1t : (M - 1);                // clamp loads
    const float* Ar0 = A + (long)ra0 * lda + kb;
    const float* Ar1 = A + (long)ra1 * lda + kb;
    const float* Bp  = B + n0 + ncol;

    v8f acc[2][4];
#pragma unroll
    for (int h = 0; h < 2; ++h)
#pragma unroll
        for (int j = 0; j < 4; ++j) acc[h][j] = v8f{};

    for (int k = 0; k < K; k += 16) {
#pragma unroll
        for (int kk = 0; kk < 16; kk += 4) {
            v2f a0 = *(const v2f*)(Ar0 + k + kk);
            v2f a1 = *(const v2f*)(Ar1 + k + kk);
            if (AMODE == 1) {   // fused BatchNorm + ReLU on A elements
                v2f sc = *(const v2f*)(ascl + k + kk + kb);
                v2f sh = *(const v2f*)(ashf + k + kk + kb);
                a0.x = fmaxf(a0.x * sc.x + sh.x, 0.0f);
                a0.y = fmaxf(a0.y * sc.y + sh.y, 0.0f);
                a1.x = fmaxf(a1.x * sc.x + sh.x, 0.0f);
                a1.y = fmaxf(a1.y * sc.y + sh.y, 0.0f);
            }
            const float* b0r = Bp + (long)(k + kk + kb) * ldb;
            const float* b1r = b0r + ldb;
#pragma unroll
            for (int j = 0; j < 4; ++j) {
                v2f b;
                b.x = b0r[j * 16];
                b.y = b1r[j * 16];
                acc[0][j] = __builtin_amdgcn_wmma_f32_16x16x4_f32(
                    false, a0, false, b, (short)0, acc[0][j], false, false);
                acc[1][j] = __builtin_amdgcn_wmma_f32_16x16x4_f32(
                    false, a1, false, b, (short)0, acc[1][j], false, false);
            }
        }
    }

    // C layout: VGPR r holds M=r (lanes 0-15) / M=r+8 (lanes 16-31), N=lane&15
    const int rhalf = (lane >> 4) << 3;
    float bj[4];
#pragma unroll
    for (int j = 0; j < 4; ++j) bj[j] = bias[n0 + j * 16 + ncol];

    float ssum[4] = {0.f, 0.f, 0.f, 0.f};
    float sqq[4]  = {0.f, 0.f, 0.f, 0.f};

#pragma unroll
    for (int h = 0; h < 2; ++h) {
#pragma unroll
        for (int r = 0; r < 8; ++r) {
            const int row = m0 + h * 16 + r + rhalf;
            if (row < M) {
                float* crow = C + (long)row * ldc + n0 + ncol;
#pragma unroll
                for (int j = 0; j < 4; ++j) {
                    float v = acc[h][j][r] + bj[j];
                    if (EPI == 0 && relu) v = fmaxf(v, 0.0f);
                    crow[j * 16] = v;
                    if (EPI == 1) { ssum[j] += v; sqq[j] += v * v; }
                }
            }
        }
    }
    if (EPI == 1) {
#pragma unroll
        for (int j = 0; j < 4; ++j) {
            unsafeAtomicAdd(&sums[n0 + j * 16 + ncol], ssum[j]);
            unsafeAtomicAdd(&sqs[n0 + j * 16 + ncol], sqq[j]);
        }
    }
}

// ---------------------------------------------------------------------------
// Scatter-add: one wave per (edge | self-loop). agg[d, 0:300] += h[s, 0:300]
// agg[d, 300:600] += (edge_attr @ We + be). Self-loops: attr = onehot(6).
// ---------------------------------------------------------------------------
__global__ __launch_bounds__(256)
void gin_scatter(const float* __restrict__ h, const int* __restrict__ ei,
                 const float* __restrict__ eattr, const float* __restrict__ We,
                 const float* __restrict__ be, float* __restrict__ agg,
                 int E, int Nn)
{
    __shared__ float sWe[7 * EMB];
    __shared__ float sbe[EMB];
    for (int i = threadIdx.x; i < 7 * EMB; i += 256) sWe[i] = We[i];
    for (int i = threadIdx.x; i < EMB; i += 256) sbe[i] = be[i];
    __syncthreads();

    const int lane = threadIdx.x & 31;
    const int wave = threadIdx.x >> 5;
    const long total = (long)E + Nn;

    for (long it = (long)blockIdx.x * 8 + wave; it < total;
         it += (long)gridDim.x * 8) {
        int s, d;
        float ea[7];
        if (it < E) {
            s = ei[it];
            d = ei[(long)E + it];
#pragma unroll
            for (int k = 0; k < 7; ++k) ea[k] = eattr[it * 7 + k];
        } else {
            s = d = (int)(it - E);
#pragma unroll
            for (int k = 0; k < 7; ++k) ea[k] = (k == 6) ? 1.0f : 0.0f;
        }
        const float* hs = h + (long)s * EMBP;
        float* ax = agg + (long)d * DP;
        for (int c = lane; c < EMB; c += 32) {
            unsafeAtomicAdd(ax + c, hs[c]);
            float ev = sbe[c];
#pragma unroll
            for (int k = 0; k < 7; ++k) ev += ea[k] * sWe[k * EMB + c];
            unsafeAtomicAdd(ax + EMB + c, ev);
        }
    }
}

// ---------------------------------------------------------------------------
// BN finalize: scale/shift from accumulated sums/sumsq
// ---------------------------------------------------------------------------
__global__ void bn_finalize(const float* __restrict__ sums,
                            const float* __restrict__ sqs,
                            const float* __restrict__ g,
                            const float* __restrict__ bt,
                            float* __restrict__ scale, float* __restrict__ shift,
                            int M, int C)
{
    int c = blockIdx.x * blockDim.x + threadIdx.x;
    if (c >= C) return;
    float inv = 1.0f / (float)M;
    float mu = sums[c] * inv;
    float var = fmaxf(sqs[c] * inv - mu * mu, 0.0f);
    float sc = g[c] * rsqrtf(var + 1e-5f);
    scale[c] = sc;
    shift[c] = bt[c] - mu * sc;
}

// ---------------------------------------------------------------------------
// Utility kernels
// ---------------------------------------------------------------------------
__global__ __launch_bounds__(256)
void zero_f32(float* __restrict__ p, long n)
{
    for (long i = (long)blockIdx.x * 256 + threadIdx.x; i < n;
         i += (long)gridDim.x * 256)
        p[i] = 0.0f;
}

// dst[dr x dc] = src[sr x sc] zero-padded
__global__ __launch_bounds__(256)
void pad_matrix(float* __restrict__ dst, const float* __restrict__ src,
                int sr, int sc, int dr, int dc)
{
    const long n = (long)dr * dc;
    for (long i = (long)blockIdx.x * 256 + threadIdx.x; i < n;
         i += (long)gridDim.x * 256) {
        int r = (int)(i / dc);
        int c = (int)(i % dc);
        dst[i] = (r < sr && c < sc) ? src[(long)r * sc + c] : 0.0f;
    }
}

__global__ __launch_bounds__(256)
void copy_out(const float* __restrict__ h, float* __restrict__ out)
{
    const float* hr = h + (long)blockIdx.x * EMBP;
    float* orow = out + (long)blockIdx.x * EMB;
    for (int c = threadIdx.x; c < EMB; c += 256) orow[c] = hr[c];
}

// ---------------------------------------------------------------------------
// Host orchestration
// ---------------------------------------------------------------------------
static inline int gemm_blocks(int M, int N)
{
    long tiles = (long)((M + 31) / 32) * (N / 64);
    return (int)((tiles + 7) / 8);
}

extern "C" void kernel_launch(void* const* d_in, const int* in_sizes, int n_in,
                              void* d_out, int out_size, void* d_ws, size_t ws_size,
                              hipStream_t stream)
{
    const float* x     = (const float*)d_in[0];   // [N, 1024]
    const int*   ei    = (const int*)  d_in[1];   // [2, E]
    const float* eattr = (const float*)d_in[2];   // [E, 7]
    const float* Win   = (const float*)d_in[3];   // [1024, 300]
    const float* bin   = (const float*)d_in[4];   // [300]
    const float* We    = (const float*)d_in[5];   // [5, 7, 300]
    const float* be    = (const float*)d_in[6];   // [5, 300]
    const float* W1    = (const float*)d_in[7];   // [5, 600, 600]
    const float* b1    = (const float*)d_in[8];   // [5, 600]
    const float* g     = (const float*)d_in[9];   // [5, 600]
    const float* bt    = (const float*)d_in[10];  // [5, 600]
    const float* W2    = (const float*)d_in[11];  // [5, 600, 300]
    const float* b2    = (const float*)d_in[12];  // [5, 300]
    float* out = (float*)d_out;

    // ---- workspace carve-up (floats) ----
    float* ws = (float*)d_ws;
    float* h     = ws; ws += (size_t)NN * EMBP;           // 16.0 M
    float* agg   = ws; ws += (size_t)NN * DP;             // 32.0 M
    float* z     = ws; ws += (size_t)NN * DP;             // 32.0 M
    float* Win_p = ws; ws += (size_t)INDIM * EMBP;        // 0.33 M
    float* W1_p  = ws; ws += (size_t)NLAYER * DP * DP;    // 2.05 M
    float* W2_p  = ws; ws += (size_t)NLAYER * DP * EMBP;  // 1.02 M
    float* bin_p = ws; ws += EMBP;
    float* b1_p  = ws; ws += NLAYER * DP;
    float* g_p   = ws; ws += NLAYER * DP;
    float* bt_p  = ws; ws += NLAYER * DP;
    float* b2_p  = ws; ws += NLAYER * EMBP;
    float* sums  = ws; ws += DP;
    float* sqs   = ws; ws += DP;   // contiguous after sums (zeroed together)
    float* scl   = ws; ws += DP;
    float* shf   = ws; ws += DP;

    // ---- pad parameters into WMMA-friendly shapes ----
    pad_matrix<<<256, 256, 0, stream>>>(Win_p, Win, INDIM, EMB, INDIM, EMBP);
    pad_matrix<<<1, 256, 0, stream>>>(bin_p, bin, 1, EMB, 1, EMBP);
    for (int l = 0; l < NLAYER; ++l) {
        pad_matrix<<<256, 256, 0, stream>>>(W1_p + (size_t)l * DP * DP,
                                            W1 + (size_t)l * D2 * D2, D2, D2, DP, DP);
        pad_matrix<<<128, 256, 0, stream>>>(W2_p + (size_t)l * DP * EMBP,
                                            W2 + (size_t)l * D2 * EMB, D2, EMB, DP, EMBP);
        pad_matrix<<<1, 256, 0, stream>>>(b1_p + l * DP, b1 + l * D2, 1, D2, 1, DP);
        pad_matrix<<<1, 256, 0, stream>>>(g_p  + l * DP, g  + l * D2, 1, D2, 1, DP);
        pad_matrix<<<1, 256, 0, stream>>>(bt_p + l * DP, bt + l * D2, 1, D2, 1, DP);
        pad_matrix<<<1, 256, 0, stream>>>(b2_p + l * EMBP, b2 + l * EMB, 1, EMB, 1, EMBP);
    }

    // ---- input projection: h = x @ Win + bin ----
    gemm_wmma_f32<0, 0><<<gemm_blocks(NN, EMBP), 256, 0, stream>>>(
        x, Win_p, bin_p, h, nullptr, nullptr, nullptr, nullptr,
        NN, EMBP, INDIM, INDIM, EMBP, EMBP, 0);

    // ---- GIN layers ----
    for (int l = 0; l < NLAYER; ++l) {
        zero_f32<<<4096, 256, 0, stream>>>(agg, (long)NN * DP);
        gin_scatter<<<2048, 256, 0, stream>>>(h, ei, eattr,
                                              We + (size_t)l * 7 * EMB,
                                              be + (size_t)l * EMB, agg, EE, NN);
        // z = agg @ W1 + b1, with fused column sum/sumsq accumulation
        zero_f32<<<1, 256, 0, stream>>>(sums, 2L * DP);
        gemm_wmma_f32<0, 1><<<gemm_blocks(NN, DP), 256, 0, stream>>>(
            agg, W1_p + (size_t)l * DP * DP, b1_p + l * DP, z,
            nullptr, nullptr, sums, sqs,
            NN, DP, DP, DP, DP, DP, 0);
        bn_finalize<<<(DP + 255) / 256, 256, 0, stream>>>(
            sums, sqs, g_p + l * DP, bt_p + l * DP, scl, shf, NN, DP);
        // h = relu(z*scale+shift) @ W2 + b2   (BN+ReLU fused into A loads;
        //  inter-layer ReLU fused into the store for l < 4)
        gemm_wmma_f32<1, 0><<<gemm_blocks(NN, EMBP), 256, 0, stream>>>(
            z, W2_p + (size_t)l * DP * EMBP, b2_p + l * EMBP, h, scl, shf,
            nullptr, nullptr,
            NN, EMBP, DP, DP, EMBP, EMBP, (l < NLAYER - 1) ? 1 : 0);
    }

    copy_out<<<NN, 256, 0, stream>>>(h, out);
}